// LinearAttentionOp_85822036509401
// MI455X (gfx1250) — compile-verified
//
#include <hip/hip_runtime.h>
#include <math.h>

typedef __attribute__((ext_vector_type(2))) float v2f;
typedef __attribute__((ext_vector_type(8))) float v8f;

// Problem constants
#define BDIM 2
#define HDIM 16
#define SDIM 2048
#define NODIM 2
#define DDIM 64
#define NCHUNK 32               // S / 64
#define TROWS 128               // 64 * NO rows per chunk
#define BH_STRIDE 262144        // S*NO*D floats per (b,h)
#define CHUNK_ELEMS 8192        // TROWS * D

// ---------------------------------------------------------------------------
// Kernel 1: per (b,h,chunk) build A = tril(k_beta k^T,-1)+I and forward-solve
// A w = k_beta, A u = v_beta (unit lower). 1024 blocks, 256 threads (8 waves).
// LDS: KC 128x68 | RHS 128x132 (cols 0-63 k_beta->w, 64-127 v_beta->u) | AM 128x132
// ---------------------------------------------------------------------------
__global__ __launch_bounds__(256) void deltanet_solve_kernel(
    const float* __restrict__ kk, const float* __restrict__ vv,
    const float* __restrict__ bb, float* __restrict__ w_ws,
    float* __restrict__ u_ws)
{
    extern __shared__ float sm[];
    float* KC  = sm;                 // 128*68  = 8704
    float* RHS = sm + 8704;          // 128*132 = 16896
    float* AM  = sm + 8704 + 16896;  // 128*132 = 16896

    const int tid = threadIdx.x;
    const int bhn = blockIdx.x;              // 0..1023
    const int n   = bhn & (NCHUNK - 1);
    const int bh  = bhn >> 5;
    const long gbase = (long)bh * BH_STRIDE + (long)n * CHUNK_ELEMS;

    const float4* k4 = (const float4*)kk + (gbase >> 2);
    const float4* v4 = (const float4*)vv + (gbase >> 2);
    const float4* b4 = (const float4*)bb + (gbase >> 2);

    // Stage: KC = k rows, RHS = [k*beta | v*beta]
    for (int i4 = tid; i4 < 2048; i4 += 256) {
        int row = i4 >> 4, e4 = i4 & 15;
        float4 kv = k4[i4];
        float4 bv = b4[i4];
        float4 vl = v4[i4];
        *(float4*)&KC[row * 68 + e4 * 4] = kv;
        float4 kb; kb.x = kv.x*bv.x; kb.y = kv.y*bv.y; kb.z = kv.z*bv.z; kb.w = kv.w*bv.w;
        float4 vb; vb.x = vl.x*bv.x; vb.y = vl.y*bv.y; vb.z = vl.z*bv.z; vb.w = vl.w*bv.w;
        int roff = row * 132 + e4 * 4;
        *(float4*)&RHS[roff]      = kb;
        *(float4*)&RHS[roff + 64] = vb;
    }
    __syncthreads();

    const int lane = tid & 31;
    const int wv   = tid >> 5;          // wave id 0..7
    const int lo   = lane & 15;
    const int hi2  = (lane >> 4) << 1;  // 0 or 2 (K offset within fragment)
    const int hi8  = (lane >> 4) << 3;  // 0 or 8 (M offset for C/D rows)

    // Phase A: AM = k_beta @ KC^T  (8x8 tiles of 16x16, K=64)
    for (int tt = wv; tt < 64; tt += 8) {
        int ti = tt >> 3, tj = tt & 7;
        v8f acc = {};
        const int arow = (ti * 16 + lo) * 132;   // k_beta row (cols 0-63)
        const int brow = (tj * 16 + lo) * 68;    // KC row (transposed B read)
        #pragma unroll
        for (int k0 = 0; k0 < 64; k0 += 4) {
            v2f a, b;
            a.x = RHS[arow + k0 + hi2];     a.y = RHS[arow + k0 + hi2 + 1];
            b.x = KC [brow + k0 + hi2];     b.y = KC [brow + k0 + hi2 + 1];
            acc = __builtin_amdgcn_wmma_f32_16x16x4_f32(false, a, false, b,
                                                        (short)0, acc, false, false);
        }
        const int crow = ti * 16 + hi8;
        const int ccol = tj * 16 + lo;
        #pragma unroll
        for (int i = 0; i < 8; ++i) AM[(crow + i) * 132 + ccol] = acc[i];
    }
    __syncthreads();

    // Blocked forward substitution over 8 blocks of 16 rows, 128 RHS cols
    for (int I = 0; I < 8; ++I) {
        if (I > 0) {
            // RHS[I] -= sum_{J<I} A[I,J] @ RHS[J]   via WMMA (K=16 per block)
            {
                const int ct = wv;       // one 16-col tile per wave
                v8f acc = {};
                for (int J = 0; J < I; ++J) {
                    const int arow = (I * 16 + lo) * 132 + J * 16;
                    #pragma unroll
                    for (int k0 = 0; k0 < 16; k0 += 4) {
                        v2f a, b;
                        a.x = AM[arow + k0 + hi2];
                        a.y = AM[arow + k0 + hi2 + 1];
                        b.x = RHS[(J * 16 + k0 + hi2)     * 132 + ct * 16 + lo];
                        b.y = RHS[(J * 16 + k0 + hi2 + 1) * 132 + ct * 16 + lo];
                        acc = __builtin_amdgcn_wmma_f32_16x16x4_f32(false, a, false, b,
                                                                    (short)0, acc, false, false);
                    }
                }
                const int crow = I * 16 + hi8;
                const int ccol = ct * 16 + lo;
                #pragma unroll
                for (int i = 0; i < 8; ++i) RHS[(crow + i) * 132 + ccol] -= acc[i];
            }
        }
        __syncthreads();
        // Diagonal 16x16 unit-lower solve: each thread owns one RHS column,
        // so the 15-step recurrence has no cross-thread dependencies.
        if (tid < 128) {
            const int col = tid;
            for (int i = 1; i < 16; ++i) {
                const int r = I * 16 + i;
                float accv = RHS[r * 132 + col];
                for (int j = 0; j < i; ++j)
                    accv -= AM[r * 132 + I * 16 + j] * RHS[(I * 16 + j) * 132 + col];
                RHS[r * 132 + col] = accv;
            }
        }
        __syncthreads();
    }

    // Write w, u to workspace (chunk-contiguous layout matching inputs)
    for (int idx = tid; idx < CHUNK_ELEMS; idx += 256) {
        int row = idx >> 6, e = idx & 63;
        w_ws[gbase + idx] = RHS[row * 132 + e];
        u_ws[gbase + idx] = RHS[row * 132 + 64 + e];
    }
}

// ---------------------------------------------------------------------------
// Kernel 2: sequential chunk scan per (b,h). 32 blocks, 256 threads.
// State collapse: M0=W0@h0, M1=W1@h0, P=Q1@h0 (WMMA), then elementwise.
// Next chunk's operands are prefetched (global_prefetch_b8) during compute.
// ---------------------------------------------------------------------------
__global__ __launch_bounds__(256) void deltanet_scan_kernel(
    const float* __restrict__ qq, const float* __restrict__ al,
    const float* __restrict__ w_ws, const float* __restrict__ u_ws,
    const int* __restrict__ linflag, float* __restrict__ out,
    float* __restrict__ hfin)
{
    extern __shared__ float sm[];
    float* H0  = sm;             // 64x68
    float* W0  = sm + 4352;
    float* W1  = sm + 8704;
    float* Q1  = sm + 13056;
    float* U0  = sm + 17408;     // becomes u_val0
    float* U1  = sm + 21760;     // becomes u_val1
    float* A0  = sm + 26112;
    float* M0  = sm + 30464;
    float* M1  = sm + 34816;     // reused as h_new
    float* P   = sm + 39168;
    float* dww = sm + 43520;     // [64]
    float* dqw = sm + 43584;     // [64]
    float* xn  = sm + 43648;     // [64]
    float* A1r = sm + 43712;     // [64]

    const int tid = threadIdx.x;
    const int bh  = blockIdx.x;
    const int lin = linflag[0];
    const long bhbase = (long)bh * BH_STRIDE;

    for (int idx = tid; idx < 4352; idx += 256) H0[idx] = 1e-6f;
    __syncthreads();

    const int lane = tid & 31;
    const int wv   = tid >> 5;          // 0..7
    const int lo   = lane & 15;
    const int hi2  = (lane >> 4) << 1;
    const int hi8  = (lane >> 4) << 3;

    for (int n = 0; n < NCHUNK; ++n) {
        const long cb  = bhbase + (long)n * CHUNK_ELEMS;
        const long cb4 = cb >> 2;
        const float4* w4 = (const float4*)w_ws + cb4;
        const float4* u4 = (const float4*)u_ws + cb4;
        const float4* q4 = (const float4*)qq + cb4;
        const float4* a4 = (const float4*)al + cb4;

        // Stage chunk: even rows -> step0, odd rows -> step1
        for (int i4 = tid; i4 < 1024; i4 += 256) {
            int c = i4 >> 4, e4 = i4 & 15;
            int off = c * 68 + e4 * 4;
            int g0 = c * 32 + e4;              // float4 index of row 2c
            *(float4*)&W0[off] = w4[g0];
            *(float4*)&W1[off] = w4[g0 + 16];
            *(float4*)&U0[off] = u4[g0];
            *(float4*)&U1[off] = u4[g0 + 16];
            *(float4*)&Q1[off] = q4[g0 + 16];
            *(float4*)&A0[off] = a4[g0];
        }
        if (tid < 16) *(float4*)&A1r[tid * 4] = a4[127 * 16 + tid];

        // Prefetch next chunk (128 B per thread per array covers 32 KB each)
        if (n + 1 < NCHUNK) {
            const long nb = bhbase + (long)(n + 1) * CHUNK_ELEMS + (long)tid * 32;
            __builtin_prefetch(w_ws + nb, 0, 3);
            __builtin_prefetch(u_ws + nb, 0, 3);
            __builtin_prefetch(qq + nb, 0, 3);
            __builtin_prefetch(al + nb, 0, 3);
        }
        __syncthreads();

        // 3 GEMMs (64x64x64): M0=W0@H0, M1=W1@H0, P=Q1@H0
        for (int tt = wv; tt < 48; tt += 8) {
            int g  = tt >> 4;
            int rr = (tt >> 2) & 3;
            int cc = tt & 3;
            const float* As = (g == 0) ? W0 : ((g == 1) ? W1 : Q1);
            float* Ds       = (g == 0) ? M0 : ((g == 1) ? M1 : P);
            v8f acc = {};
            const int arow = (rr * 16 + lo) * 68;
            #pragma unroll
            for (int k0 = 0; k0 < 64; k0 += 4) {
                v2f a, b;
                a.x = As[arow + k0 + hi2];  a.y = As[arow + k0 + hi2 + 1];
                b.x = H0[(k0 + hi2)     * 68 + cc * 16 + lo];
                b.y = H0[(k0 + hi2 + 1) * 68 + cc * 16 + lo];
                acc = __builtin_amdgcn_wmma_f32_16x16x4_f32(false, a, false, b,
                                                            (short)0, acc, false, false);
            }
            const int crow = rr * 16 + hi8;
            const int ccol = cc * 16 + lo;
            #pragma unroll
            for (int i = 0; i < 8; ++i) Ds[(crow + i) * 68 + ccol] = acc[i];
        }
        __syncthreads();

        // u_val0 = U0 - M0 ; row dots (disjoint buffers, same phase)
        for (int idx = tid; idx < 4096; idx += 256) {
            int c = idx >> 6, e = idx & 63;
            U0[c * 68 + e] -= M0[c * 68 + e];
        }
        if (tid < 64) {
            float s1 = 0.f, s2 = 0.f;
            const int r = tid * 68;
            for (int e = 0; e < 64; ++e) {
                float w0v = W0[r + e];
                s1 += W1[r + e] * w0v;
                s2 += Q1[r + e] * w0v;
            }
            dww[tid] = s1;
            dqw[tid] = s2;
        }
        __syncthreads();

        // u_val1 and output o
        float* outp = out + (long)bh * (SDIM * DDIM) + (long)n * 4096;
        for (int idx = tid; idx < 4096; idx += 256) {
            int c = idx >> 6, e = idx & 63;
            int o68 = c * 68 + e;
            float a0v  = A0[o68];
            float uv0v = U0[o68];
            float uv1  = U1[o68] - (M1[o68] * a0v + dww[c] * uv0v);
            U1[o68] = uv1;
            outp[idx] = Q1[o68] * uv1 + P[o68] * a0v + dqw[c] * uv0v;
        }
        __syncthreads();

        // h_new (only position c=63 matters), into M1
        for (int idx = tid; idx < 4096; idx += 256) {
            int r = idx >> 6, e = idx & 63;
            float hv = (H0[r * 68 + e] * A0[63 * 68 + e] +
                        W0[63 * 68 + r] * U0[63 * 68 + e]) * A1r[e] +
                       W1[63 * 68 + r] * U1[63 * 68 + e];
            M1[r * 68 + e] = hv;
        }
        __syncthreads();

        if (!lin) {
            if (tid < 64) {
                float s = 0.f;
                for (int e = 0; e < 64; ++e) {
                    float xv = M1[tid * 68 + e];
                    s += xv * xv;
                }
                xn[tid] = sqrtf(s) + 1e-6f;
            }
            __syncthreads();
            for (int idx = tid; idx < 4096; idx += 256) {
                int r = idx >> 6, e = idx & 63;
                float xv = M1[r * 68 + e];
                float y  = 2.f * xv / xn[r];
                float g  = 0.5f * y * (1.f + tanhf(0.7978845608028654f *
                                                   (y + 0.044715f * y * y * y)));
                H0[r * 68 + e] = 0.5f * xv * g;
            }
        } else {
            for (int idx = tid; idx < 4096; idx += 256) {
                int r = idx >> 6, e = idx & 63;
                H0[r * 68 + e] = M1[r * 68 + e];
            }
        }
        __syncthreads();
    }

    // h_final
    float* hf = hfin + (long)bh * 4096;
    for (int idx = tid; idx < 4096; idx += 256) {
        int r = idx >> 6, e = idx & 63;
        hf[idx] = H0[r * 68 + e];
    }
}

// ---------------------------------------------------------------------------
extern "C" void kernel_launch(void* const* d_in, const int* in_sizes, int n_in,
                              void* d_out, int out_size, void* d_ws, size_t ws_size,
                              hipStream_t stream)
{
    const float* q     = (const float*)d_in[0];
    const float* k     = (const float*)d_in[1];
    const float* v     = (const float*)d_in[2];
    const float* alpha = (const float*)d_in[3];
    const float* beta  = (const float*)d_in[4];
    const int*   lin   = (const int*)d_in[5];

    float* out  = (float*)d_out;
    float* hfin = out + (long)BDIM * HDIM * SDIM * DDIM;   // 4,194,304 floats in

    float* w_ws = (float*)d_ws;                            // 32 MB
    float* u_ws = w_ws + (long)BDIM * HDIM * SDIM * NODIM * DDIM; // +32 MB

    const int smem1 = (8704 + 16896 + 16896) * 4;          // 169,984 B
    const int smem2 = 43776 * 4;                           // 175,104 B
    (void)hipFuncSetAttribute((const void*)deltanet_solve_kernel,
                              hipFuncAttributeMaxDynamicSharedMemorySize, smem1);
    (void)hipFuncSetAttribute((const void*)deltanet_scan_kernel,
                              hipFuncAttributeMaxDynamicSharedMemorySize, smem2);

    deltanet_solve_kernel<<<BDIM * HDIM * NCHUNK, 256, smem1, stream>>>(
        k, v, beta, w_ws, u_ws);
    deltanet_scan_kernel<<<BDIM * HDIM, 256, smem2, stream>>>(
        q, alpha, w_ws, u_ws, lin, out, hfin);
}